// GINBlock_19576460935444
// MI455X (gfx1250) — compile-verified
//
#include <hip/hip_runtime.h>

typedef __attribute__((ext_vector_type(2))) float v2f;
typedef __attribute__((ext_vector_type(8))) float v8f;

#define BN_EPS 1e-5f

static __device__ __forceinline__ v8f wmma4(v2f a, v2f b, v8f c) {
    return __builtin_amdgcn_wmma_f32_16x16x4_f32(false, a, false, b, (short)0, c,
                                                 false, false);
}

// ---------------------------------------------------------------------------
// Kernel 1: zero the aggregation buffer + BN stat accumulators
// ---------------------------------------------------------------------------
__global__ __launch_bounds__(256) void zero_ws_kernel(float* __restrict__ agg,
                                                      size_t n_agg,
                                                      float* __restrict__ stats) {
    size_t i = (size_t)blockIdx.x * blockDim.x + threadIdx.x;
    size_t stride = (size_t)gridDim.x * blockDim.x;
    for (size_t j = i; j < n_agg; j += stride) agg[j] = 0.0f;
    if (i < 512) stats[i] = 0.0f;   // sum[256] + sumsq[256]
}

// ---------------------------------------------------------------------------
// Kernel 2: GIN neighbor aggregation. One wave32 per edge: lane l moves
// float4 at column 4*l (coalesced 512B gather), then 4 native f32 atomics
// into agg[dst]. x/agg stay L2-resident (51 MB each vs 192 MB L2).
// ---------------------------------------------------------------------------
__global__ __launch_bounds__(256) void edge_agg_kernel(const float* __restrict__ x,
                                                       const long long* __restrict__ ei,
                                                       float* __restrict__ agg,
                                                       long long E) {
    const int lane = threadIdx.x & 31;
    const long long wid = (long long)blockIdx.x * (blockDim.x >> 5) + (threadIdx.x >> 5);
    if (wid >= E) return;
    const long long src = ei[wid];
    const long long dst = ei[E + wid];
    const float4 v = *(const float4*)&x[src * 128 + lane * 4];
    float* a = &agg[dst * 128 + lane * 4];
    (void)__hip_atomic_fetch_add(&a[0], v.x, __ATOMIC_RELAXED, __HIP_MEMORY_SCOPE_AGENT);
    (void)__hip_atomic_fetch_add(&a[1], v.y, __ATOMIC_RELAXED, __HIP_MEMORY_SCOPE_AGENT);
    (void)__hip_atomic_fetch_add(&a[2], v.z, __ATOMIC_RELAXED, __HIP_MEMORY_SCOPE_AGENT);
    (void)__hip_atomic_fetch_add(&a[3], v.w, __ATOMIC_RELAXED, __HIP_MEMORY_SCOPE_AGENT);
}

// ---------------------------------------------------------------------------
// Kernel 3: h1 = ((1+eps)*x + agg) @ W1 + b1, plus BN sum/sumsq reduction.
// Block = 256 threads = 8 waves, 32-row M stripe. Wave w owns n-tiles
// {w, w+8} x m-tiles {0,1}: 2x2 register tiles -> 4 WMMAs per
// (2 LDS A-frags + 4 global B loads). K=128 in steps of 4.
// ---------------------------------------------------------------------------
__global__ __launch_bounds__(256) void gin_gemm1_kernel(const float* __restrict__ x,
                                                        const float* __restrict__ agg,
                                                        const float* __restrict__ eps_p,
                                                        const float* __restrict__ W1,
                                                        const float* __restrict__ b1,
                                                        float* __restrict__ h1,
                                                        float* __restrict__ sum_g,
                                                        float* __restrict__ sq_g,
                                                        int Nn) {
    __shared__ __align__(16) float As[32 * 132];  // stride 132 (==4 mod 64): conflict-free frags
    __shared__ float sumS[256];
    __shared__ float sqS[256];

    const int tid  = threadIdx.x;
    const int row0 = blockIdx.x * 32;
    const bool full = (row0 + 32) <= Nn;          // uniform: tail never taken for N=100000
    const float ep = 1.0f + eps_p[0];

    sumS[tid] = 0.0f;
    sqS[tid]  = 0.0f;

    // Stage A tile: 32 rows x 128 cols of (1+eps)*x + agg. 1024 float4 / 256 thr.
    #pragma unroll
    for (int i = 0; i < 4; ++i) {
        const int f   = tid + i * 256;
        const int r   = f >> 5;           // 0..31
        const int c   = (f & 31) * 4;     // 0..124
        const int row = row0 + r;
        float4 o = float4{0, 0, 0, 0};
        if (full || row < Nn) {
            const float4 xv = *(const float4*)&x[(size_t)row * 128 + c];
            const float4 gv = *(const float4*)&agg[(size_t)row * 128 + c];
            o.x = fmaf(ep, xv.x, gv.x); o.y = fmaf(ep, xv.y, gv.y);
            o.z = fmaf(ep, xv.z, gv.z); o.w = fmaf(ep, xv.w, gv.w);
        }
        *(float4*)&As[r * 132 + c] = o;
    }
    __syncthreads();

    const int lane = tid & 31;
    const int wv   = tid >> 5;      // 0..7
    const int half = lane >> 4;     // K sub-offset for fragments
    const int l16  = lane & 15;
    const int nc0  = wv * 16 + l16;         // n-tile w
    const int nc1  = 128 + wv * 16 + l16;   // n-tile w+8

    v8f acc[2][2] = {{{}, {}}, {{}, {}}};   // [m-tile][n-tile]

    #pragma unroll 2
    for (int k0 = 0; k0 < 128; k0 += 4) {
        const int ka = k0 + half * 2;
        const v2f a0 = *(const v2f*)&As[l16 * 132 + ka];         // rows 0..15
        const v2f a1 = *(const v2f*)&As[(16 + l16) * 132 + ka];  // rows 16..31
        v2f bf0, bf1;
        bf0.x = W1[(size_t)(ka + 0) * 256 + nc0];
        bf0.y = W1[(size_t)(ka + 1) * 256 + nc0];
        bf1.x = W1[(size_t)(ka + 0) * 256 + nc1];
        bf1.y = W1[(size_t)(ka + 1) * 256 + nc1];
        acc[0][0] = wmma4(a0, bf0, acc[0][0]);
        acc[0][1] = wmma4(a0, bf1, acc[0][1]);
        acc[1][0] = wmma4(a1, bf0, acc[1][0]);
        acc[1][1] = wmma4(a1, bf1, acc[1][1]);
    }

    #pragma unroll
    for (int mi = 0; mi < 2; ++mi) {
        #pragma unroll
        for (int ni = 0; ni < 2; ++ni) {
            const int ncol = ni ? nc1 : nc0;
            const float bias = b1[ncol];
            const int rbase = row0 + mi * 16 + half * 8;
            float s = 0.0f, q = 0.0f;
            if (full) {
                #pragma unroll
                for (int i = 0; i < 8; ++i) {
                    const float v = acc[mi][ni][i] + bias;
                    h1[(size_t)(rbase + i) * 256 + ncol] = v;
                    s += v; q += v * v;
                }
            } else {
                #pragma unroll
                for (int i = 0; i < 8; ++i) {
                    const float v = acc[mi][ni][i] + bias;
                    if (rbase + i < Nn) { h1[(size_t)(rbase + i) * 256 + ncol] = v;
                                          s += v; q += v * v; }
                }
            }
            (void)__hip_atomic_fetch_add(&sumS[ncol], s, __ATOMIC_RELAXED,
                                         __HIP_MEMORY_SCOPE_WORKGROUP);
            (void)__hip_atomic_fetch_add(&sqS[ncol], q, __ATOMIC_RELAXED,
                                         __HIP_MEMORY_SCOPE_WORKGROUP);
        }
    }
    __syncthreads();
    // one global atomic per column per block
    (void)__hip_atomic_fetch_add(&sum_g[tid], sumS[tid], __ATOMIC_RELAXED, __HIP_MEMORY_SCOPE_AGENT);
    (void)__hip_atomic_fetch_add(&sq_g[tid],  sqS[tid],  __ATOMIC_RELAXED, __HIP_MEMORY_SCOPE_AGENT);
}

// ---------------------------------------------------------------------------
// Kernel 4: fold BN stats into per-column scale/shift
// ---------------------------------------------------------------------------
__global__ __launch_bounds__(256) void bn_finalize_kernel(const float* __restrict__ sum_g,
                                                          const float* __restrict__ sq_g,
                                                          const float* __restrict__ gamma,
                                                          const float* __restrict__ beta,
                                                          float* __restrict__ scale,
                                                          float* __restrict__ shift,
                                                          float invN) {
    const int c = threadIdx.x;
    const float mu  = sum_g[c] * invN;
    const float var = fmaxf(sq_g[c] * invN - mu * mu, 0.0f);
    const float sc  = gamma[c] * rsqrtf(var + BN_EPS);
    scale[c] = sc;
    shift[c] = beta[c] - mu * sc;
}

// ---------------------------------------------------------------------------
// Kernel 5: out = relu(h1*scale + shift) @ W2 + b2. BN+ReLU fused into the
// LDS A-tile staging. 32-row stripe, wave w owns n-tile w x m-tiles {0,1}.
// ---------------------------------------------------------------------------
__global__ __launch_bounds__(256) void gin_gemm2_kernel(const float* __restrict__ h1,
                                                        const float* __restrict__ scale,
                                                        const float* __restrict__ shift,
                                                        const float* __restrict__ W2,
                                                        const float* __restrict__ b2,
                                                        float* __restrict__ out,
                                                        int Nn) {
    __shared__ __align__(16) float As[32 * 260];  // stride 260 (==4 mod 64): conflict-free

    const int tid  = threadIdx.x;
    const int row0 = blockIdx.x * 32;
    const bool full = (row0 + 32) <= Nn;

    // Stage A tile: 32 rows x 256 cols; BN scale/shift + ReLU applied on the fly.
    #pragma unroll
    for (int i = 0; i < 8; ++i) {
        const int f   = tid + i * 256;
        const int r   = f >> 6;           // 0..31
        const int c   = (f & 63) * 4;     // 0..252
        const int row = row0 + r;
        float4 o = float4{0, 0, 0, 0};
        if (full || row < Nn) {
            const float4 hv = *(const float4*)&h1[(size_t)row * 256 + c];
            const float4 sc = *(const float4*)&scale[c];
            const float4 sh = *(const float4*)&shift[c];
            o.x = fmaxf(0.0f, fmaf(hv.x, sc.x, sh.x));
            o.y = fmaxf(0.0f, fmaf(hv.y, sc.y, sh.y));
            o.z = fmaxf(0.0f, fmaf(hv.z, sc.z, sh.z));
            o.w = fmaxf(0.0f, fmaf(hv.w, sc.w, sh.w));
        }
        *(float4*)&As[r * 260 + c] = o;
    }
    __syncthreads();

    const int lane = tid & 31;
    const int wv   = tid >> 5;      // n-tile 0..7 (covers 128 output cols)
    const int half = lane >> 4;
    const int l16  = lane & 15;
    const int ncol = wv * 16 + l16;

    v8f acc[2] = {{}, {}};          // [m-tile]

    #pragma unroll 2
    for (int k0 = 0; k0 < 256; k0 += 4) {
        const int ka = k0 + half * 2;
        const v2f a0 = *(const v2f*)&As[l16 * 260 + ka];
        const v2f a1 = *(const v2f*)&As[(16 + l16) * 260 + ka];
        v2f bf;
        bf.x = W2[(size_t)(ka + 0) * 128 + ncol];
        bf.y = W2[(size_t)(ka + 1) * 128 + ncol];
        acc[0] = wmma4(a0, bf, acc[0]);
        acc[1] = wmma4(a1, bf, acc[1]);
    }

    const float bias = b2[ncol];
    #pragma unroll
    for (int mi = 0; mi < 2; ++mi) {
        const int rbase = row0 + mi * 16 + half * 8;
        if (full) {
            #pragma unroll
            for (int i = 0; i < 8; ++i)
                out[(size_t)(rbase + i) * 128 + ncol] = acc[mi][i] + bias;
        } else {
            #pragma unroll
            for (int i = 0; i < 8; ++i)
                if (rbase + i < Nn)
                    out[(size_t)(rbase + i) * 128 + ncol] = acc[mi][i] + bias;
        }
    }
}

// ---------------------------------------------------------------------------
extern "C" void kernel_launch(void* const* d_in, const int* in_sizes, int n_in,
                              void* d_out, int out_size, void* d_ws, size_t ws_size,
                              hipStream_t stream) {
    const float*     x     = (const float*)d_in[0];
    const long long* ei    = (const long long*)d_in[1];   // int64 edge_index (2,E)
    const float*     W1    = (const float*)d_in[2];
    const float*     b1    = (const float*)d_in[3];
    const float*     gamma = (const float*)d_in[4];
    const float*     beta  = (const float*)d_in[5];
    const float*     W2    = (const float*)d_in[6];
    const float*     b2    = (const float*)d_in[7];
    const float*     eps   = (const float*)d_in[8];
    float*           out   = (float*)d_out;

    const long long N = in_sizes[0] / 128;
    const long long E = in_sizes[1] / 2;

    // Workspace layout (fp32): agg[N*128] | h1[N*256] | sum[256] sq[256] scale[256] shift[256]
    float* ws    = (float*)d_ws;
    float* agg   = ws;
    float* h1    = agg + (size_t)N * 128;
    float* stats = h1 + (size_t)N * 256;
    float* sum_g = stats;
    float* sq_g  = stats + 256;
    float* scale = stats + 512;
    float* shift = stats + 768;

    zero_ws_kernel<<<2048, 256, 0, stream>>>(agg, (size_t)N * 128, stats);

    const unsigned eblocks = (unsigned)((E + 7) / 8);   // 8 waves/block, 1 edge/wave
    edge_agg_kernel<<<eblocks, 256, 0, stream>>>(x, ei, agg, E);

    const unsigned mblocks = (unsigned)((N + 31) / 32);
    gin_gemm1_kernel<<<mblocks, 256, 0, stream>>>(x, agg, eps, W1, b1, h1, sum_g, sq_g, (int)N);

    bn_finalize_kernel<<<1, 256, 0, stream>>>(sum_g, sq_g, gamma, beta, scale, shift,
                                              1.0f / (float)N);

    gin_gemm2_kernel<<<mblocks, 256, 0, stream>>>(h1, scale, shift, W2, b2, out, (int)N);
}